// WeightedGraphNet_14998025797985
// MI455X (gfx1250) — compile-verified
//
#include <hip/hip_runtime.h>
#include <math.h>

#define NN    10000      // nodes
#define NE    160000     // edges
#define FN    8
#define FE    4
#define C     24
#define ITERS 4
#define HID   16
#define WCOLS 144        // 96 edge-weight cols | 24 edge-bias cols | 24 root cols
#define HEAD_BLOCKS 240
#define EPB   128        // edges per block in edge_scatter (NE % EPB == 0)
#define MT    5          // M-tiles per wave in node GEMM (625 % 5 == 0)

typedef __attribute__((ext_vector_type(2))) float v2f;
typedef __attribute__((ext_vector_type(8))) float v8f;

#define NEG_INF __int_as_float(0xff800000)

// ---------------------------------------------------------------------------
// Deterministic float atomic max via signed/unsigned integer ordering trick.
// ---------------------------------------------------------------------------
__device__ __forceinline__ void atomicMaxF32(float* addr, float val) {
    if (val >= 0.0f) {
        atomicMax((int*)addr, __float_as_int(val));
    } else {
        atomicMin((unsigned int*)addr, (unsigned int)__float_as_uint(val));
    }
}

// ---------------------------------------------------------------------------
// Build fused per-layer weight matrices W' [Kdim x 144]:
//   cols [0,96)   : edge_w[f, i*C+o]   (f = col/24, o = col%24)
//   cols [96,120) : edge_b[i*C+o]
//   cols [120,144): root[i, o]
// ---------------------------------------------------------------------------
__global__ void prep_weights(const float* __restrict__ few, const float* __restrict__ feb,
                             const float* __restrict__ ew,  const float* __restrict__ eb,
                             const float* __restrict__ root0, const float* __restrict__ roots,
                             float* __restrict__ W0, float* __restrict__ Wit) {
    int stride = gridDim.x * blockDim.x;
    for (int idx = blockIdx.x * blockDim.x + threadIdx.x; idx < FN * WCOLS; idx += stride) {
        int i = idx / WCOLS, cidx = idx % WCOLS;
        float v;
        if (cidx < 96)       { int f = cidx / C, o = cidx % C; v = few[f * (FN * C) + i * C + o]; }
        else if (cidx < 120) { v = feb[i * C + (cidx - 96)]; }
        else                 { v = root0[i * C + (cidx - 120)]; }
        W0[idx] = v;
    }
    for (int idx = blockIdx.x * blockDim.x + threadIdx.x; idx < ITERS * C * WCOLS; idx += stride) {
        int it = idx / (C * WCOLS);
        int r  = idx % (C * WCOLS);
        int i = r / WCOLS, cidx = r % WCOLS;
        float v;
        if (cidx < 96)       { int f = cidx / C, o = cidx % C; v = ew[f * (C * C) + i * C + o]; }
        else if (cidx < 120) { v = eb[i * C + (cidx - 96)]; }
        else                 { v = roots[it * C * C + i * C + (cidx - 120)]; }
        Wit[idx] = v;
    }
}

// ---------------------------------------------------------------------------
// Degree
// ---------------------------------------------------------------------------
__global__ void zero_deg(int* __restrict__ deg) {
    int t = blockIdx.x * blockDim.x + threadIdx.x;
    if (t < NN) deg[t] = 0;
}
__global__ void count_deg(const int* __restrict__ dst, int* __restrict__ deg) {
    int t = blockIdx.x * blockDim.x + threadIdx.x;
    if (t < NE) atomicAdd(&deg[dst[t]], 1);
}
__global__ void init_agg(float* __restrict__ agg) {
    int t = blockIdx.x * blockDim.x + threadIdx.x;
    if (t < NN * C) agg[t] = NEG_INF;
}

// ---------------------------------------------------------------------------
// Node GEMM:  T[10000 x 144] = A[10000 x KD] @ Wp[KD x 144]
// One wave per (column-strip, 5 M-tiles). B strip held in registers across
// the 5 tiles; KD/4 chained V_WMMA_F32_16X16X4_F32 per tile. All tile dims
// divide exactly -> no tails, EXEC all-ones at every WMMA.
// A layout (16x4 f32): lanes 0-15 carry K=k0,k0+1; lanes 16-31 carry K=k0+2,k0+3.
// ---------------------------------------------------------------------------
template <int KD>
__global__ __launch_bounds__(32) void node_gemm_wmma(const float* __restrict__ A,
                                                     const float* __restrict__ Wp,
                                                     float* __restrict__ T) {
    constexpr int KSTEPS = KD / 4;
    const int tiles_n = WCOLS / 16;                 // 9
    int tn   = blockIdx.x % tiles_n;
    int tm0  = (blockIdx.x / tiles_n) * MT;
    int lane = threadIdx.x;
    int half = lane >> 4;
    int l    = lane & 15;
    int colB = tn * 16 + l;

    v2f b[KSTEPS];
#pragma unroll
    for (int s = 0; s < KSTEPS; ++s) {
        int ka = 4 * s + half * 2;
        b[s].x = Wp[ka * WCOLS + colB];
        b[s].y = Wp[(ka + 1) * WCOLS + colB];
    }

#pragma unroll
    for (int mt = 0; mt < MT; ++mt) {
        int rowA = (tm0 + mt) * 16 + l;
        v8f acc = {};
#pragma unroll
        for (int s = 0; s < KSTEPS; ++s) {
            int ka = 4 * s + half * 2;
            v2f a;
            a.x = A[rowA * KD + ka];
            a.y = A[rowA * KD + ka + 1];
            acc = __builtin_amdgcn_wmma_f32_16x16x4_f32(
                false, a, false, b[s], (short)0, acc, false, false);
        }
        int baseRow = (tm0 + mt) * 16 + half * 8;
#pragma unroll
        for (int i = 0; i < 8; ++i)
            T[(size_t)(baseRow + i) * WCOLS + tn * 16 + l] = acc[i];
    }
}

// ---------------------------------------------------------------------------
// Edge scatter: 128 edges/block; per-edge metadata staged once in LDS; each
// work item covers 4 channels via float4 (b128) loads of the gathered T row;
// exact (deterministic) atomic max into agg[dst].
// ---------------------------------------------------------------------------
__global__ __launch_bounds__(256) void edge_scatter(const float* __restrict__ T,
                                                    const float* __restrict__ ea,
                                                    const int* __restrict__ src,
                                                    const int* __restrict__ dst,
                                                    float* __restrict__ agg) {
    __shared__ float4 s_ea[EPB];
    __shared__ int    s_src[EPB];
    __shared__ int    s_dst[EPB];
    int e0  = blockIdx.x * EPB;
    int tid = threadIdx.x;
    if (tid < EPB) {
        int e = e0 + tid;
        s_ea[tid]  = *(const float4*)(ea + (size_t)e * FE);
        s_src[tid] = src[e];
        s_dst[tid] = dst[e];
    }
    __syncthreads();

    // EPB edges * 6 channel-quads = 768 items / 256 threads = 3 passes
#pragma unroll
    for (int it = 0; it < (EPB * 6) / 256; ++it) {
        int i  = it * 256 + tid;
        int el = i / 6;
        int j  = (i % 6) * 4;                       // channel quad base
        float4 a = s_ea[el];
        const float* Ts = T + (size_t)s_src[el] * WCOLS + j;
        float4 t0 = *(const float4*)(Ts);
        float4 t1 = *(const float4*)(Ts + 24);
        float4 t2 = *(const float4*)(Ts + 48);
        float4 t3 = *(const float4*)(Ts + 72);
        float4 tb = *(const float4*)(Ts + 96);
        float4 m;
        m.x = tb.x + a.x * t0.x + a.y * t1.x + a.z * t2.x + a.w * t3.x;
        m.y = tb.y + a.x * t0.y + a.y * t1.y + a.z * t2.y + a.w * t3.y;
        m.z = tb.z + a.x * t0.z + a.y * t1.z + a.z * t2.z + a.w * t3.z;
        m.w = tb.w + a.x * t0.w + a.y * t1.w + a.z * t2.w + a.w * t3.w;
        float* ag = agg + (size_t)s_dst[el] * C + j;
        atomicMaxF32(ag + 0, m.x);
        atomicMaxF32(ag + 1, m.y);
        atomicMaxF32(ag + 2, m.z);
        atomicMaxF32(ag + 3, m.w);
    }
}

// ---------------------------------------------------------------------------
// h = relu6( (deg>0 ? agg : 0) + root-part-of-T + bias ); also resets agg to
// -inf for the next layer's segment-max (fused re-init).
// ---------------------------------------------------------------------------
__global__ __launch_bounds__(256) void combine(float* __restrict__ agg,
                                               const float* __restrict__ T,
                                               const int* __restrict__ deg,
                                               const float* __restrict__ bias,
                                               float* __restrict__ h) {
    int t = blockIdx.x * blockDim.x + threadIdx.x;
    if (t >= NN * C) return;
    int n = t / C, o = t % C;
    float a = (deg[n] > 0) ? agg[t] : 0.0f;
    agg[t] = NEG_INF;                               // ready for next layer
    float v = a + T[(size_t)n * WCOLS + 120 + o] + bias[o];
    h[t] = fminf(fmaxf(v, 0.0f), 6.0f);
}

// ---------------------------------------------------------------------------
// Head, stage 1: per-block partial sums of flat(h) @ w1  (w1: [240000 x 16])
// ---------------------------------------------------------------------------
__global__ __launch_bounds__(256) void head_partial(const float* __restrict__ h,
                                                    const float* __restrict__ w1,
                                                    float* __restrict__ partial) {
    float loc[HID];
#pragma unroll
    for (int j = 0; j < HID; ++j) loc[j] = 0.0f;

    int stride = gridDim.x * blockDim.x;
    for (int idx = blockIdx.x * blockDim.x + threadIdx.x; idx < NN * C; idx += stride) {
        float f = h[idx];
        const float4* wr = (const float4*)(w1 + (size_t)idx * HID);
        float4 w0 = wr[0], wv1 = wr[1], wv2 = wr[2], wv3 = wr[3];
        loc[0]  += f * w0.x;  loc[1]  += f * w0.y;  loc[2]  += f * w0.z;  loc[3]  += f * w0.w;
        loc[4]  += f * wv1.x; loc[5]  += f * wv1.y; loc[6]  += f * wv1.z; loc[7]  += f * wv1.w;
        loc[8]  += f * wv2.x; loc[9]  += f * wv2.y; loc[10] += f * wv2.z; loc[11] += f * wv2.w;
        loc[12] += f * wv3.x; loc[13] += f * wv3.y; loc[14] += f * wv3.z; loc[15] += f * wv3.w;
    }

    __shared__ float sh[8][HID];
    int lane = threadIdx.x & 31, wv = threadIdx.x >> 5;
#pragma unroll
    for (int j = 0; j < HID; ++j) {
        float v = loc[j];
        for (int off = 16; off > 0; off >>= 1) v += __shfl_down(v, off, 32);
        if (lane == 0) sh[wv][j] = v;
    }
    __syncthreads();
    if (threadIdx.x < HID) {
        float s = 0.0f;
#pragma unroll
        for (int w = 0; w < 8; ++w) s += sh[w][threadIdx.x];
        partial[blockIdx.x * HID + threadIdx.x] = s;
    }
}

// ---------------------------------------------------------------------------
// Head, stage 2: reduce partials, ELU, dot with w2, ELU -> out[0]
// ---------------------------------------------------------------------------
__global__ __launch_bounds__(32) void head_final(const float* __restrict__ partial, int nblocks,
                                                 const float* __restrict__ b1,
                                                 const float* __restrict__ w2,
                                                 const float* __restrict__ b2,
                                                 float* __restrict__ out) {
    float s = 0.0f;
    int j = threadIdx.x;
    if (j < HID) {
        for (int b = 0; b < nblocks; ++b) s += partial[b * HID + j];
        s += b1[j];
        s = (s > 0.0f) ? s : expm1f(s);   // ELU
        s *= w2[j];
    }
    for (int off = 16; off > 0; off >>= 1) s += __shfl_down(s, off, 32);
    if (threadIdx.x == 0) {
        float o = s + b2[0];
        out[0] = (o > 0.0f) ? o : expm1f(o);
    }
}

// ---------------------------------------------------------------------------
// Launch
// ---------------------------------------------------------------------------
extern "C" void kernel_launch(void* const* d_in, const int* in_sizes, int n_in,
                              void* d_out, int out_size, void* d_ws, size_t ws_size,
                              hipStream_t stream) {
    const float* x    = (const float*)d_in[0];
    const float* ea   = (const float*)d_in[1];
    const int*   src  = (const int*)d_in[2];
    const int*   dst  = src + NE;
    const float* few  = (const float*)d_in[3];
    const float* feb  = (const float*)d_in[4];
    const float* ew   = (const float*)d_in[5];
    const float* eb   = (const float*)d_in[6];
    const float* root0  = (const float*)d_in[7];
    const float* bias0  = (const float*)d_in[8];
    const float* roots  = (const float*)d_in[9];
    const float* biases = (const float*)d_in[10];
    const float* w1   = (const float*)d_in[11];
    const float* b1   = (const float*)d_in[12];
    const float* w2   = (const float*)d_in[13];
    const float* b2   = (const float*)d_in[14];
    float* out = (float*)d_out;

    // workspace carve-up (floats)
    float* W0  = (float*)d_ws;               // 8*144
    float* Wit = W0  + FN * WCOLS;           // 4*24*144
    float* T   = Wit + ITERS * C * WCOLS;    // 10000*144
    float* agg = T   + (size_t)NN * WCOLS;   // 10000*24
    float* h   = agg + NN * C;               // 10000*24
    int*   deg = (int*)(h + NN * C);         // 10000
    float* partial = (float*)(deg + NN);     // HEAD_BLOCKS*16

    const int tilesM = NN / 16, tilesN = WCOLS / 16;      // 625 x 9
    dim3 gemmGrid((tilesM / MT) * tilesN), gemmBlk(32);   // 1125 waves
    int aggBlocks  = (NN * C + 255) / 256;
    int edgeBlocks = NE / EPB;                            // 1250

    prep_weights<<<64, 256, 0, stream>>>(few, feb, ew, eb, root0, roots, W0, Wit);
    zero_deg<<<(NN + 255) / 256, 256, 0, stream>>>(deg);
    count_deg<<<(NE + 255) / 256, 256, 0, stream>>>(dst, deg);
    init_agg<<<aggBlocks, 256, 0, stream>>>(agg);

    // layer 0: K = FN = 8
    node_gemm_wmma<FN><<<gemmGrid, gemmBlk, 0, stream>>>(x, W0, T);
    edge_scatter<<<edgeBlocks, 256, 0, stream>>>(T, ea, src, dst, agg);
    combine<<<aggBlocks, 256, 0, stream>>>(agg, T, deg, bias0, h);

    // 4 iteration layers: K = C = 24 (combine re-arms agg each time)
    for (int it = 0; it < ITERS; ++it) {
        node_gemm_wmma<C><<<gemmGrid, gemmBlk, 0, stream>>>(h, Wit + (size_t)it * C * WCOLS, T);
        edge_scatter<<<edgeBlocks, 256, 0, stream>>>(T, ea, src, dst, agg);
        combine<<<aggBlocks, 256, 0, stream>>>(agg, T, deg, biases + it * C, h);
    }

    // output head
    head_partial<<<HEAD_BLOCKS, 256, 0, stream>>>(h, w1, partial);
    head_final<<<1, 32, 0, stream>>>(partial, HEAD_BLOCKS, b1, w2, b2, out);
}